// BistableBank_68418829025680
// MI455X (gfx1250) — compile-verified
//
#include <hip/hip_runtime.h>

typedef __attribute__((ext_vector_type(2))) float v2f;
typedef __attribute__((ext_vector_type(4))) float v4f;
typedef __attribute__((ext_vector_type(8))) float v8f;

#define MU_MIN 0.1f
#define MU_MAX 1.5f

__device__ __forceinline__ float sigmoidf(float x) {
    return 1.0f / (1.0f + __expf(-x));
}

// W[m][i][j] = sigmoid(0.5*(g[m,i,j]+g[m,j,i])) with zero diagonal
__device__ __forceinline__ float Wmij(const float* __restrict__ g, int m, int i, int j) {
    if (i == j) return 0.0f;
    float s = 0.5f * (g[(m * 4 + i) * 4 + j] + g[(m * 4 + j) * 4 + i]);
    return sigmoidf(s);
}

// ---------------------------------------------------------------------------
// Main kernel: one wave processes 16 items per tile via V_WMMA_F32_16X16X4_F32
//   coup_all (16 items x 16 (m,i)) = A (16x4 of X) * B (4x16 stacked W)
// Epilogue routes per-item mode z and finishes the elementwise update.
// ---------------------------------------------------------------------------
__global__ __launch_bounds__(256) void bistable_wmma_kernel(
    const float* __restrict__ X, const int* __restrict__ z,
    const float* __restrict__ dt_val, const float* __restrict__ g,
    const float* __restrict__ mu_logits, const float* __restrict__ alpha_p,
    const float* __restrict__ beta_p, float* __restrict__ out, int nTiles)
{
    __shared__ float s_mu[16];   // mu[m*4+i]
    __shared__ float s_deg[16];  // deg[m*4+i]
    __shared__ float s_a[4];
    __shared__ float s_b[4];
    __shared__ __align__(16) float s_coup[8][16][16]; // [wave][n=(4m+i)][item]

    const int tid  = threadIdx.x;
    const int lane = tid & 31;
    const int wave = tid >> 5;

    // ---- one-time per-block parameter tables -------------------------------
    if (tid < 16) {
        const int m = tid >> 2, i = tid & 3;
        s_mu[tid] = MU_MIN + (MU_MAX - MU_MIN) * sigmoidf(mu_logits[tid]);
        float d = 0.0f;
        #pragma unroll
        for (int j = 0; j < 4; ++j) d += Wmij(g, m, i, j);
        s_deg[tid] = d;
    }
    if (tid < 4) { s_a[tid] = alpha_p[tid]; s_b[tid] = beta_p[tid]; }
    __syncthreads();

    // ---- loop-invariant B fragment (4x16, f32, 2 VGPRs) --------------------
    // lane layout (mirrors ISA 32-bit 16x4 A layout): lanes 0-15 -> K=0/1,
    // lanes 16-31 -> K=2/3; column n = lane%16 maps to (m = n>>2, i = n&3).
    const int n  = lane & 15;
    const int bm = n >> 2, bi = n & 3;
    const int k0 = (lane < 16) ? 0 : 2;
    v2f Bfrag;
    Bfrag.x = Wmij(g, bm, bi, k0);
    Bfrag.y = Wmij(g, bm, bi, k0 + 1);

    const float dt = dt_val[0];

    const int wavesPerGrid = (gridDim.x * blockDim.x) >> 5;
    const int gwave = (blockIdx.x * blockDim.x + tid) >> 5;

    for (int tile = gwave; tile < nTiles; tile += wavesPerGrid) {
        const long base = (long)tile * 16; // first item of this wave's tile

        // A fragment: lane holds X[item = n][k0 .. k0+1]  (coalesced 8B loads)
        v2f Afrag = *(const v2f*)(X + (base + n) * 4 + k0);

        v8f acc = {};
        // D = A(16x4 f32) x B(4x16 f32) + 0   -> v_wmma_f32_16x16x4_f32
        acc = __builtin_amdgcn_wmma_f32_16x16x4_f32(
            false, Afrag, false, Bfrag, (short)0, acc, false, false);

        // Stash D into per-wave LDS tile: VGPR r holds (item = r + (lane<16?0:8), col n).
        // Layout [n][item] makes the 8 accumulator regs two contiguous b128 stores.
        {
            const int ibase = (lane < 16) ? 0 : 8;
            float* dst = &s_coup[wave][n][ibase];
            *(v4f*)(dst)     = v4f{acc[0], acc[1], acc[2], acc[3]};
            *(v4f*)(dst + 4) = v4f{acc[4], acc[5], acc[6], acc[7]};
        }
        // LDS ops are in-order within a wave; fence compiler + hardware completion.
        asm volatile("s_wait_dscnt 0" ::: "memory");

        // ---- epilogue: lane handles item = lane>>1, nodes j0, j0+1 ---------
        const int item = lane >> 1;
        const int j0   = (lane & 1) * 2;
        const int zm   = z[base + item];
        const v2f xv   = *(const v2f*)(X + (base + item) * 4 + j0);
        const float a  = s_a[zm];
        const float b  = s_b[zm];

        float o0, o1;
        {
            const float xj   = xv.x;
            const float coup = s_coup[wave][zm * 4 + j0][item];
            const float loc  = s_mu[zm * 4 + j0] * xj - xj * xj * xj;
            const float cpl  = a * s_deg[zm * 4 + j0] * xj + b * coup;
            o0 = xj + dt * (loc + cpl);
        }
        {
            const float xj   = xv.y;
            const float coup = s_coup[wave][zm * 4 + j0 + 1][item];
            const float loc  = s_mu[zm * 4 + j0 + 1] * xj - xj * xj * xj;
            const float cpl  = a * s_deg[zm * 4 + j0 + 1] * xj + b * coup;
            o1 = xj + dt * (loc + cpl);
        }
        *(v2f*)(out + (base + item) * 4 + j0) = v2f{o0, o1};

        // WAR fence before next iteration's LDS stores into the same tile.
        asm volatile("s_wait_dscnt 0" ::: "memory");
    }
}

// ---------------------------------------------------------------------------
// Scalar tail for B % 16 items (not hit for B = 4,194,304; kept for safety).
// ---------------------------------------------------------------------------
__global__ void bistable_tail_kernel(
    const float* __restrict__ X, const int* __restrict__ z,
    const float* __restrict__ dt_val, const float* __restrict__ g,
    const float* __restrict__ mu_logits, const float* __restrict__ alpha_p,
    const float* __restrict__ beta_p, float* __restrict__ out,
    long start, long nItems)
{
    const long r = blockIdx.x * (long)blockDim.x + threadIdx.x;
    if (r >= nItems) return;
    const long idx = start + r;
    const int m = z[idx];
    const float dt = dt_val[0];
    float x[4];
    #pragma unroll
    for (int j = 0; j < 4; ++j) x[j] = X[idx * 4 + j];
    #pragma unroll
    for (int i = 0; i < 4; ++i) {
        float coup = 0.0f, d = 0.0f;
        #pragma unroll
        for (int j = 0; j < 4; ++j) {
            const float w = Wmij(g, m, i, j);
            coup += w * x[j];
            d += w;
        }
        const float mu  = MU_MIN + (MU_MAX - MU_MIN) * sigmoidf(mu_logits[m * 4 + i]);
        const float loc = mu * x[i] - x[i] * x[i] * x[i];
        const float cpl = alpha_p[m] * d * x[i] + beta_p[m] * coup;
        out[idx * 4 + i] = x[i] + dt * (loc + cpl);
    }
}

extern "C" void kernel_launch(void* const* d_in, const int* in_sizes, int n_in,
                              void* d_out, int out_size, void* d_ws, size_t ws_size,
                              hipStream_t stream) {
    const float* X   = (const float*)d_in[0];
    const int*   z   = (const int*)d_in[1];
    const float* dt  = (const float*)d_in[2];
    const float* g   = (const float*)d_in[3];
    const float* mu  = (const float*)d_in[4];
    const float* a   = (const float*)d_in[5];
    const float* b   = (const float*)d_in[6];
    float* out = (float*)d_out;

    const long B      = in_sizes[0] / 4; // X is (B, 4)
    const long nTiles = B / 16;
    const long rem    = B % 16;

    if (nTiles > 0) {
        // 8 waves/block (wave32). 4096 blocks -> 32768 waves -> 8 tiles/wave at B=4.19M.
        long grid = (nTiles + 7) / 8;     // aim ~8 tiles per wave min
        grid = (grid + 7) / 8;            // 8 waves per block
        if (grid > 4096) grid = 4096;
        if (grid < 1) grid = 1;
        bistable_wmma_kernel<<<(int)grid, 256, 0, stream>>>(
            X, z, dt, g, mu, a, b, out, (int)nTiles);
    }
    if (rem > 0) {
        bistable_tail_kernel<<<(int)((rem + 255) / 256), 256, 0, stream>>>(
            X, z, dt, g, mu, a, b, out, nTiles * 16, rem);
    }
}